// AlgoMiniBatch_541165879939
// MI455X (gfx1250) — compile-verified
//
#include <hip/hip_runtime.h>
#include <math.h>

typedef __attribute__((ext_vector_type(2))) float v2f;
typedef __attribute__((ext_vector_type(8))) float v8f;

#define D_FEAT   256
#define HIDDEN   256
#define KDIM     512      // 2*D_FEAT == 2*HIDDEN
#define BATCH    1024
#define S1       25
#define S2       10
#define M1       (BATCH * (1 + S2))   // 11264 rows through layer-1 GEMM
#define LDS_STRIDE 516                // k-stage stride: 512+4 pad (16B aligned, kills bank conflicts)
#define Z_STRIDE   260                // epilogue tile stride: 256+4 pad (16B aligned)

// XOR-butterfly lane exchange via ds_swizzle_b32 (group-of-32 mode):
// offset = {1'b0, xor_mask[4:0], or_mask[4:0]=0, and_mask[4:0]=0x1F}
// mask must be an integer constant expression -> template parameter.
template <int MASK>
__device__ __forceinline__ float swz_xor(float v)
{
    return __int_as_float(
        __builtin_amdgcn_ds_swizzle(__float_as_int(v), (MASK << 10) | 0x1F));
}

// ---------------------------------------------------------------------------
// K1: cat1[r, 0:256] = x[self], cat1[r, 256:512] = mean_{j<25} x[nb[j]]
//     r < 1024          -> self = targets[r],  nb = nb1_self[r, :]
//     r in [1024,11264) -> q = r-1024, self = nb2[q], nb = nb1_nb[q, :]
// 256-thread block = 4 rows x 64 lanes; float4 per lane (global_load_b128)
// ---------------------------------------------------------------------------
__global__ __launch_bounds__(256) void gather_cat1(
    const float* __restrict__ x,
    const int*   __restrict__ targets,
    const int*   __restrict__ nb1_self,
    const int*   __restrict__ nb2,
    const int*   __restrict__ nb1_nb,
    float*       __restrict__ cat1)
{
    const int tid = threadIdx.x;
    const int sub = tid >> 6;            // row within block (0..3)
    const int c4  = tid & 63;            // float4 channel (0..63)
    const int r   = blockIdx.x * 4 + sub;

    int self;
    const int* nb;
    if (r < BATCH) {
        self = targets[r];
        nb   = nb1_self + r * S1;
    } else {
        const int q = r - BATCH;
        self = nb2[q];
        nb   = nb1_nb + q * S1;
    }

    const float4* xv    = reinterpret_cast<const float4*>(x);
    float4*       cat1v = reinterpret_cast<float4*>(cat1);

    // self half
    cat1v[(size_t)r * 128 + c4] = xv[(size_t)self * 64 + c4];

    // mean over 25 neighbors
    float4 s = make_float4(0.f, 0.f, 0.f, 0.f);
    #pragma unroll
    for (int j = 0; j < S1; ++j) {
        const float4 t = xv[(size_t)nb[j] * 64 + c4];
        s.x += t.x; s.y += t.y; s.z += t.z; s.w += t.w;
    }
    const float inv = 1.0f / (float)S1;
    s.x *= inv; s.y *= inv; s.z *= inv; s.w *= inv;
    cat1v[(size_t)r * 128 + 64 + c4] = s;
}

// ---------------------------------------------------------------------------
// K2/K4: out[m, :] = L2normalize( relu( A[m,:] @ W^T + bias ) )  (N=256, K=512)
// block = 512 threads = 16 waves; block owns a 16-row M-tile and ALL 256 cols
// (wave w -> N-tile 16*w), so bias+relu+row-L2-norm fuse entirely in-block.
// f32 WMMA 16x16x4, 128 K-steps. Epilogue bounces tile through LDS for
// coalesced b128 stores.
// ---------------------------------------------------------------------------
__global__ __launch_bounds__(512) void sage_gemm_relu_norm(
    const float* __restrict__ A,      // [M, 512]
    const float* __restrict__ W,      // [256, 512] row-major (N, K)
    const float* __restrict__ bias,   // [256]
    float*       __restrict__ out)    // [M, 256]
{
    __shared__ float As[16 * LDS_STRIDE];   // ~33 KB; reused for output tile
    __shared__ float ssq[16][16];           // [row][wave] partial sum-of-squares
    __shared__ float rinv[16];              // per-row 1/norm

    const int tid  = threadIdx.x;
    const int lane = tid & 31;
    const int wave = tid >> 5;              // 0..15 -> N tile
    const int m0   = blockIdx.x * 16;

    // ---- stage A tile: 16 rows x 512 floats = 2048 float4, 4 per thread ----
    {
        const float4* Ag = reinterpret_cast<const float4*>(A + (size_t)m0 * KDIM);
        #pragma unroll
        for (int i = 0; i < 4; ++i) {
            const int idx = tid + i * 512;      // float4 index, 128 per row
            const int row = idx >> 7;
            const int col = idx & 127;
            *reinterpret_cast<float4*>(&As[row * LDS_STRIDE + col * 4]) = Ag[idx];
        }
    }
    __syncthreads();

    const int half = lane >> 4;             // 0: K {0,1}, 1: K {2,3}
    const int l15  = lane & 15;
    const int n0   = wave * 16;
    const int koff = half * 2;

    const float* Arow = As + l15 * LDS_STRIDE + koff;
    const float* Wrow = W + (size_t)(n0 + l15) * KDIM + koff;

    v8f acc = {};
    #pragma unroll 8
    for (int k = 0; k < KDIM; k += 4) {
        v2f a = *reinterpret_cast<const v2f*>(Arow + k);   // ds_load
        v2f b = *reinterpret_cast<const v2f*>(Wrow + k);   // global_load_b64
        acc = __builtin_amdgcn_wmma_f32_16x16x4_f32(
            false, a, false, b, (short)0, acc, false, false);
    }

    // ---- bias + relu; per-row sum of squares (16 cols of this wave) ----
    // C layout: VGPR i -> row = half*8 + i, col = n0 + l15
    const float bv = bias[n0 + l15];
    float vals[8];
    #pragma unroll
    for (int i = 0; i < 8; ++i) {
        float v = acc[i] + bv;
        v = v > 0.0f ? v : 0.0f;
        vals[i] = v;

        // XOR butterfly over each 16-lane half via ds_swizzle_b32:
        // masks 1,2,4,8 never cross the 16-lane boundary; all lanes end with
        // the half's sum of squares.
        float s = v * v;
        s += swz_xor<1>(s);
        s += swz_xor<2>(s);
        s += swz_xor<4>(s);
        s += swz_xor<8>(s);
        if (l15 == 0) ssq[half * 8 + i][wave] = s;
    }
    __syncthreads();   // also guarantees all As reads are done (reuse below)

    // ---- cross-wave reduction -> per-row inverse norm ----
    if (tid < 16) {
        float t = 0.0f;
        #pragma unroll
        for (int w = 0; w < 16; ++w) t += ssq[tid][w];
        rinv[tid] = (t > 0.0f) ? (1.0f / sqrtf(t)) : 0.0f;
    }
    __syncthreads();

    // ---- scale and deposit tile into LDS (stride 260, conflict-free) ----
    {
        const int row0 = half * 8;
        #pragma unroll
        for (int i = 0; i < 8; ++i) {
            const int row = row0 + i;
            As[row * Z_STRIDE + n0 + l15] = vals[i] * rinv[row];
        }
    }
    __syncthreads();

    // ---- cooperative coalesced store: 16 rows x 64 float4 = 1024 vec4 ----
    {
        float4* ov = reinterpret_cast<float4*>(out + (size_t)m0 * HIDDEN);
        #pragma unroll
        for (int i = 0; i < 2; ++i) {
            const int idx = tid + i * 512;      // 0..1023
            const int row = idx >> 6;
            const int col = idx & 63;           // float4 column
            ov[row * 64 + col] =
                *reinterpret_cast<const float4*>(&As[row * Z_STRIDE + col * 4]);
        }
    }
}

// ---------------------------------------------------------------------------
// K3: cat2[b, 0:256] = h1[b, :], cat2[b, 256:512] = mean_j h1[1024 + b*10 + j, :]
// 256-thread block = 4 batch rows x 64 float4 lanes
// ---------------------------------------------------------------------------
__global__ __launch_bounds__(256) void build_cat2(
    const float* __restrict__ h1,     // [11264, 256]: rows 0..1023 = h1_t, rest = h1_n
    float*       __restrict__ cat2)   // [1024, 512]
{
    const int tid = threadIdx.x;
    const int sub = tid >> 6;
    const int c4  = tid & 63;
    const int b   = blockIdx.x * 4 + sub;

    const float4* hv = reinterpret_cast<const float4*>(h1);
    float4*       cv = reinterpret_cast<float4*>(cat2);

    cv[(size_t)b * 128 + c4] = hv[(size_t)b * 64 + c4];

    float4 s = make_float4(0.f, 0.f, 0.f, 0.f);
    const float4* hn = hv + (size_t)(BATCH + b * S2) * 64 + c4;
    #pragma unroll
    for (int j = 0; j < S2; ++j) {
        const float4 t = hn[(size_t)j * 64];
        s.x += t.x; s.y += t.y; s.z += t.z; s.w += t.w;
    }
    const float inv = 1.0f / (float)S2;
    s.x *= inv; s.y *= inv; s.z *= inv; s.w *= inv;
    cv[(size_t)b * 128 + 64 + c4] = s;
}

// ---------------------------------------------------------------------------
extern "C" void kernel_launch(void* const* d_in, const int* in_sizes, int n_in,
                              void* d_out, int out_size, void* d_ws, size_t ws_size,
                              hipStream_t stream)
{
    const float* x        = (const float*)d_in[0];
    const int*   targets  = (const int*)  d_in[1];
    const int*   nb1_self = (const int*)  d_in[2];
    const int*   nb2      = (const int*)  d_in[3];
    const int*   nb1_nb   = (const int*)  d_in[4];
    const float* W1       = (const float*)d_in[5];
    const float* b1       = (const float*)d_in[6];
    const float* W2       = (const float*)d_in[7];
    const float* b2       = (const float*)d_in[8];
    float* out = (float*)d_out;

    // workspace layout (floats): ~36.7 MB total
    float* cat1 = (float*)d_ws;                         // 11264*512
    float* h1   = cat1 + (size_t)M1 * KDIM;             // 11264*256
    float* cat2 = h1   + (size_t)M1 * HIDDEN;           // 1024*512

    // layer 1: 11264 rows (1024 targets + 10240 layer-2 neighbors), norm fused
    gather_cat1<<<M1 / 4, 256, 0, stream>>>(x, targets, nb1_self, nb2, nb1_nb, cat1);
    sage_gemm_relu_norm<<<M1 / 16, 512, 0, stream>>>(cat1, W1, b1, h1);

    // layer 2
    build_cat2<<<BATCH / 4, 256, 0, stream>>>(h1, cat2);
    sage_gemm_relu_norm<<<BATCH / 16, 512, 0, stream>>>(cat2, W2, b2, out);
}